// NR_GraphAttention_601295421712
// MI455X (gfx1250) — compile-verified
//
#include <hip/hip_runtime.h>
#include <stdint.h>

#define NODE_N 100000
#define REL_N  2000
#define EDGE_N 600000
#define DIM    128
#define OUTC   384          // 3 * DIM, concat of the 3 layer outputs
#define WPB    4            // waves per block (wave32)
#define TPB    (WPB * 32)
#define BATCH  8            // rows per TDM gather (32-bit index mode max)

typedef __attribute__((ext_vector_type(4))) unsigned int u32x4;
typedef __attribute__((ext_vector_type(8))) int i32x8;
typedef __attribute__((ext_vector_type(4))) int i32x4;

__device__ __forceinline__ float dot4(float4 a, float4 b) {
    return fmaf(a.x, b.x, fmaf(a.y, b.y, fmaf(a.z, b.z, a.w * b.w)));
}

// wave32 all-lanes sum reduction
__device__ __forceinline__ float waveRedSum(float x) {
    x += __shfl_xor(x, 1, 32);
    x += __shfl_xor(x, 2, 32);
    x += __shfl_xor(x, 4, 32);
    x += __shfl_xor(x, 8, 32);
    x += __shfl_xor(x, 16, 32);
    return x;
}

__device__ __forceinline__ int lowerBound(const int* __restrict__ a, int n, int key) {
    int lo = 0, hi = n;
    while (lo < hi) {
        int mid = (lo + hi) >> 1;
        if (a[mid] < key) lo = mid + 1; else hi = mid;
    }
    return lo;
}

// TDM gather-mode load: fetch `n` (<=8) indexed rows of `rowLen` f32 elements
// from a row-major table (row stride `rowStride` elems) into LDS at `ldsByte`.
// Rows land contiguously: row i at ldsByte + i*rowLen*4. (CDNA5 ISA)
__device__ __forceinline__ void tdm_gather_rows(
    uint32_t ldsByte, const float* basePtr, uint32_t numRows,
    uint32_t rowStride, int n, const int* idx)
{
    uint64_t ga = (uint64_t)(uintptr_t)basePtr;
    // D# group 0: count=1 | gather_index_size=32b | gather_mode=1, lds, gaddr, type=2
    u32x4 g0 = { 0xC0000001u,
                 ldsByte,
                 (uint32_t)ga,
                 (uint32_t)((ga >> 32) & 0x01FFFFFFu) | (2u << 30) };
    // D# group 1: data_size=4B; tensor_dim0=DIM; tensor_dim1=numRows;
    // tile_dim0=DIM; tile_dim1=n valid indices; tensor_dim0_stride=rowStride
    i32x8 g1 = { (int)(2u << 16),
                 (int)(((uint32_t)DIM & 0xFFFFu) << 16),
                 (int)(((uint32_t)DIM >> 16) | ((numRows & 0xFFFFu) << 16)),
                 (int)(((numRows >> 16) & 0xFFFFu) | (((uint32_t)DIM & 0xFFFFu) << 16)),
                 n,
                 (int)rowStride,
                 0, 0 };
    // D# groups 2/3: 32-bit row indices 0..7
    i32x4 g2 = { idx[0], idx[1], idx[2], idx[3] };
    i32x4 g3 = { idx[4], idx[5], idx[6], idx[7] };
#if __clang_major__ >= 23
    i32x8 gz = { 0, 0, 0, 0, 0, 0, 0, 0 };
    __builtin_amdgcn_tensor_load_to_lds(g0, g1, g2, g3, gz, 0);
#else
    __builtin_amdgcn_tensor_load_to_lds(g0, g1, g2, g3, 0);
#endif
}

__global__ void relu_embed_kernel(const float4* __restrict__ f, float* __restrict__ out) {
    int i = blockIdx.x * blockDim.x + threadIdx.x;  // over NODE_N * 32 float4s
    if (i >= NODE_N * (DIM / 4)) return;
    int n = i >> 5, c4 = i & 31;
    float4 v = f[i];
    float4 o;
    o.x = fmaxf(v.x, 0.f); o.y = fmaxf(v.y, 0.f);
    o.z = fmaxf(v.z, 0.f); o.w = fmaxf(v.w, 0.f);
    *(float4*)(out + (size_t)n * OUTC + c4 * 4) = o;
}

__global__ __launch_bounds__(TPB) void gat_layer_kernel(
    float* __restrict__ outbuf,          // [NODE, 384]; cols inCol..+127 = input feats
    int inCol, int outCol,
    const float* __restrict__ relEmb,    // [REL, 128]
    const float* __restrict__ spVal,     // [E]
    const int*   __restrict__ spRel,     // [E]
    const int*   __restrict__ adjSrc,    // [E], sorted ascending
    const int*   __restrict__ adjDst,    // [E]
    const float* __restrict__ ak)        // [384] attention kernel for this layer
{
    // per-wave: BATCH rel rows + BATCH neighbor rows, 512B each => 8KB/wave
    __shared__ __align__(16) float sbuf[WPB * 2 * BATCH * DIM];
    const int lane = threadIdx.x & 31;
    const int wave = threadIdx.x >> 5;
    const int v = blockIdx.x * WPB + wave;
    if (v >= NODE_N) return;

    float* relT   = &sbuf[wave * (2 * BATCH * DIM)];
    float* neighT = relT + BATCH * DIM;
    // generic LDS addresses carry the workgroup-relative offset in addr[31:0]
    const uint32_t ldsRel   = (uint32_t)(uintptr_t)relT;
    const uint32_t ldsNeigh = ldsRel + (uint32_t)(BATCH * DIM * 4);
    const int c0 = lane * 4;

    const float* inBase = outbuf + inCol;

    float4 a0 = *(const float4*)(ak + c0);
    float4 a1 = *(const float4*)(ak + DIM + c0);
    float4 a2 = *(const float4*)(ak + 2 * DIM + c0);

    // contiguous edge range of this node (adj_src sorted)
    int e0 = lowerBound(adjSrc, EDGE_N, v);
    int e1 = lowerBound(adjSrc, EDGE_N, v + 1);

    // selfs @ ak[:D] is constant per node
    float4 s4 = *(const float4*)(inBase + (size_t)v * OUTC + c0);
    float sdot = waveRedSum(dot4(s4, a0));

    // online segment softmax state
    float m = -3.402823e38f, ssum = 0.f;
    float ax = 0.f, ay = 0.f, az = 0.f, aw = 0.f;

    for (int base = e0; base < e1; base += BATCH) {
        int n = e1 - base; if (n > BATCH) n = BATCH;
        int ridx[BATCH], nidx[BATCH];
#pragma unroll
        for (int i = 0; i < BATCH; ++i) {
            bool ok = (base + i) < e1;
            ridx[i] = ok ? spRel[base + i]  : 0;
            nidx[i] = ok ? adjDst[base + i] : 0;
        }
        // one TDM op per tile: DMA engine gathers n indexed 512B rows into LDS
        tdm_gather_rows(ldsRel,   relEmb, (uint32_t)REL_N,  (uint32_t)DIM,  n, ridx);
        tdm_gather_rows(ldsNeigh, inBase, (uint32_t)NODE_N, (uint32_t)OUTC, n, nidx);
        __builtin_amdgcn_s_wait_tensorcnt(0);
        asm volatile("" ::: "memory");   // order LDS reads after the tensorcnt wait

        for (int i = 0; i < n; ++i) {
            float val = spVal[base + i];
            float4 r4 = *(const float4*)(relT   + i * DIM + c0);
            float4 n4 = *(const float4*)(neighT + i * DIM + c0);

            // rels = l2_normalize(sp_val * rel_emb[sp_rel])
            r4.x *= val; r4.y *= val; r4.z *= val; r4.w *= val;
            float rr = waveRedSum(dot4(r4, r4));
            float rn = rsqrtf(fmaxf(rr, 1e-12f));
            r4.x *= rn; r4.y *= rn; r4.z *= rn; r4.w *= rn;

            // Householder reflection: n' = n - 2 (n.r) r
            float p = 2.f * waveRedSum(dot4(n4, r4));
            n4.x = fmaf(-p, r4.x, n4.x);
            n4.y = fmaf(-p, r4.y, n4.y);
            n4.z = fmaf(-p, r4.z, n4.z);
            n4.w = fmaf(-p, r4.w, n4.w);

            // attention logit
            float att = sdot + waveRedSum(dot4(n4, a1) + dot4(r4, a2));

            // online softmax update
            float mn = fmaxf(m, att);
            float scale = __expf(m - mn);   // first edge: exp(-huge) == 0
            float w = __expf(att - mn);
            ssum = fmaf(ssum, scale, w);
            ax = fmaf(ax, scale, w * n4.x);
            ay = fmaf(ay, scale, w * n4.y);
            az = fmaf(az, scale, w * n4.z);
            aw = fmaf(aw, scale, w * n4.w);
            m = mn;
        }
    }
    float inv = (e1 > e0) ? (1.f / ssum) : 0.f; // empty segment -> 0
    float4 o;
    o.x = fmaxf(ax * inv, 0.f);  // fused ReLU
    o.y = fmaxf(ay * inv, 0.f);
    o.z = fmaxf(az * inv, 0.f);
    o.w = fmaxf(aw * inv, 0.f);
    *(float4*)(outbuf + (size_t)v * OUTC + outCol + c0) = o;
}

extern "C" void kernel_launch(void* const* d_in, const int* in_sizes, int n_in,
                              void* d_out, int out_size, void* d_ws, size_t ws_size,
                              hipStream_t stream) {
    (void)in_sizes; (void)n_in; (void)out_size; (void)d_ws; (void)ws_size;
    const float* features = (const float*)d_in[0];
    const float* rel_emb  = (const float*)d_in[1];
    const float* sp_val   = (const float*)d_in[2];
    const float* attn     = (const float*)d_in[3];   // [2, 1, 384]
    const int*   adj_src  = (const int*)d_in[4];
    const int*   adj_dst  = (const int*)d_in[5];
    const int*   sp_rel   = (const int*)d_in[6];
    // d_in[7] = sp_row == arange(E): that segment_sum is an identity gather.
    float* out = (float*)d_out;

    relu_embed_kernel<<<(NODE_N * (DIM / 4) + 255) / 256, 256, 0, stream>>>(
        (const float4*)features, out);
    for (int l = 0; l < 2; ++l) {
        gat_layer_kernel<<<(NODE_N + WPB - 1) / WPB, TPB, 0, stream>>>(
            out, l * DIM, (l + 1) * DIM,
            rel_emb, sp_val, sp_rel, adj_src, adj_dst,
            attn + (size_t)l * 3 * DIM);
    }
}